// GAT_layer_32401233281690
// MI455X (gfx1250) — compile-verified
//
#include <hip/hip_runtime.h>
#include <hip/hip_bf16.h>
#include <math.h>

// GAT layer for MI455X (gfx1250, wave32).
// Pipeline: fp32 WMMA projection GEMM -> per-node logits -> edge segment-max
// -> edge segment-sum(exp) -> weighted scatter-add -> head-mean + LayerNorm.
// All intermediate state lives in L2 (~105MB << 192MB L2), so the random
// gather/scatter edge phase never hits HBM.

typedef __attribute__((ext_vector_type(2))) float v2f;
typedef __attribute__((ext_vector_type(8))) float v8f;

#define IN_F     128
#define OUT_F    64
#define HEADS    2
#define XDIM     128          // HEADS * OUT_F
#define NEG_SLOPE 0.2f
#define LN_EPS    1e-5f
#define SM_EPS    1e-16f

// ---------------------------------------------------------------------------
// Kernel 1: x = X @ W^T with V_WMMA_F32_16X16X4_F32.
// One wave owns a 16-row strip of X and computes ALL 8 16-col output tiles
// (8 accumulators = 64 VGPRs). Per K-step: one A-fragment load feeds 8 WMMAs,
// so X is read exactly once (vs 8x in a one-tile-per-wave scheme); the 8
// B-fragment loads come from the 64KB W matrix, which stays WGP$/L2 resident.
// A (16x4 fp32) per-lane layout (ISA 7.12.2): lanes 0-15 = M rows, VGPR0/1 =
// K0/K1; lanes 16-31 = K2/K3 -> contiguous K pair == one float2 (b64) load.
// B (4x16) mirrors this with lanes indexing the N column; B[k][n] = W[n][k],
// so the B fragment is a contiguous float2 from W's row (c0+n).
// ---------------------------------------------------------------------------
__global__ __launch_bounds__(256) void gat_gemm_wmma(
    const float* __restrict__ X, const float* __restrict__ W,
    float* __restrict__ xp, int N)
{
    const int wave = (blockIdx.x * blockDim.x + threadIdx.x) >> 5;
    const int lane = threadIdx.x & 31;
    const int r0 = wave * 16;            // 16 rows of X per wave
    if (r0 >= N) return;

    const int lm = lane & 15;
    const int kk = (lane >> 4) << 1;     // 0 or 2

    int ar = r0 + lm; if (ar >= N) ar = N - 1;   // clamp partial tail tile
    const float* arow = X + (size_t)ar * IN_F + kk;
    const float* brow = W + (size_t)lm * IN_F + kk;  // + t*16*IN_F per col tile

    v8f acc[8];
    #pragma unroll
    for (int t = 0; t < 8; ++t) acc[t] = (v8f){};

    #pragma unroll 4
    for (int k0 = 0; k0 < IN_F; k0 += 4) {
        v2f a = *(const v2f*)(arow + k0);
        #pragma unroll
        for (int t = 0; t < 8; ++t) {
            v2f b = *(const v2f*)(brow + (size_t)t * 16 * IN_F + k0);
            acc[t] = __builtin_amdgcn_wmma_f32_16x16x4_f32(
                /*neg_a=*/false, a, /*neg_b=*/false, b,
                /*c_mod=*/(short)0, acc[t], /*reuse_a=*/false, /*reuse_b=*/false);
        }
    }

    // C/D layout: VGPR v -> row M = v + (lane>=16 ? 8 : 0), col = t*16 + lm
    const int mofs = (lane >> 4) << 3;
    #pragma unroll
    for (int v = 0; v < 8; ++v) {
        const int row = r0 + mofs + v;
        if (row < N) {
            float* orow = xp + (size_t)row * XDIM + lm;
            #pragma unroll
            for (int t = 0; t < 8; ++t) orow[t * 16] = acc[t][v];
        }
    }
}

// ---------------------------------------------------------------------------
// Kernel 2: per-node attention logits a_src/a_dst, plus init of m / denom /
// accumulator (workspace is poisoned by the harness, so we must zero it).
// One wave per node; lanes 0-15 = head 0, lanes 16-31 = head 1.
// ---------------------------------------------------------------------------
__global__ __launch_bounds__(256) void gat_logits_init(
    const float* __restrict__ xp,
    const float* __restrict__ attS, const float* __restrict__ attD,
    float* __restrict__ aS, float* __restrict__ aD,
    float* __restrict__ mmax, float* __restrict__ denom,
    float* __restrict__ acc, int N)
{
    const int wave = (blockIdx.x * blockDim.x + threadIdx.x) >> 5;
    const int lane = threadIdx.x & 31;
    if (wave >= N) return;
    const int head = lane >> 4;
    const int fo   = lane * 4;                      // == head*64 + (lane&15)*4

    float4 xv = *(const float4*)(xp + (size_t)wave * XDIM + fo);
    float4 sv = *(const float4*)(attS + fo);
    float4 dv = *(const float4*)(attD + fo);
    float ps = xv.x*sv.x + xv.y*sv.y + xv.z*sv.z + xv.w*sv.w;
    float pd = xv.x*dv.x + xv.y*dv.y + xv.z*dv.z + xv.w*dv.w;
    #pragma unroll
    for (int off = 8; off >= 1; off >>= 1) {        // reduce within 16-lane half
        ps += __shfl_xor(ps, off, 32);
        pd += __shfl_xor(pd, off, 32);
    }
    if ((lane & 15) == 0) {
        aS[wave * 2 + head]    = ps;
        aD[wave * 2 + head]    = pd;
        mmax[wave * 2 + head]  = -INFINITY;
        denom[wave * 2 + head] = 0.0f;
    }
    float4 z = {0.f, 0.f, 0.f, 0.f};
    *(float4*)(acc + (size_t)wave * XDIM + fo) = z;
}

// sign-split float atomic max: positive values via signed-int max, negative
// values via unsigned-int min (monotone bit encodings). Init must be -inf.
__device__ __forceinline__ void atomicMaxF(float* addr, float val)
{
    if (val >= 0.0f) atomicMax((int*)addr, __float_as_int(val));
    else             atomicMin((unsigned int*)addr, __float_as_uint(val));
}

__device__ __forceinline__ float lrelu(float v)
{
    return v > 0.0f ? v : NEG_SLOPE * v;
}

// ---------------------------------------------------------------------------
// Kernel 3: segment max of edge logits over dst. One thread per edge (both
// heads). Edges [0,E) come from edge_index; [E, E+N) are the self-loops.
// ---------------------------------------------------------------------------
__global__ __launch_bounds__(256) void gat_edge_max(
    const int* __restrict__ esrc, const int* __restrict__ edst,
    const float* __restrict__ aS, const float* __restrict__ aD,
    float* __restrict__ mmax, int E, int N)
{
    const int eid = blockIdx.x * blockDim.x + threadIdx.x;
    if (eid >= E + N) return;
    int s, d;
    if (eid < E) { s = esrc[eid]; d = edst[eid]; }
    else         { s = d = eid - E; }
    const float e0 = lrelu(aS[s * 2 + 0] + aD[d * 2 + 0]);
    const float e1 = lrelu(aS[s * 2 + 1] + aD[d * 2 + 1]);
    atomicMaxF(&mmax[d * 2 + 0], e0);
    atomicMaxF(&mmax[d * 2 + 1], e1);
}

// ---------------------------------------------------------------------------
// Kernel 4: segment sum of exp(e - m[dst]).
// ---------------------------------------------------------------------------
__global__ __launch_bounds__(256) void gat_edge_sum(
    const int* __restrict__ esrc, const int* __restrict__ edst,
    const float* __restrict__ aS, const float* __restrict__ aD,
    const float* __restrict__ mmax, float* __restrict__ denom, int E, int N)
{
    const int eid = blockIdx.x * blockDim.x + threadIdx.x;
    if (eid >= E + N) return;
    int s, d;
    if (eid < E) { s = esrc[eid]; d = edst[eid]; }
    else         { s = d = eid - E; }
    const float e0 = lrelu(aS[s * 2 + 0] + aD[d * 2 + 0]);
    const float e1 = lrelu(aS[s * 2 + 1] + aD[d * 2 + 1]);
    atomicAdd(&denom[d * 2 + 0], __expf(e0 - mmax[d * 2 + 0]));
    atomicAdd(&denom[d * 2 + 1], __expf(e1 - mmax[d * 2 + 1]));
}

// ---------------------------------------------------------------------------
// Kernel 5: weighted scatter-add of source features. One wave per edge:
// coalesced float4 gather of x[src] (512B/wave, L2-resident) then 4
// GLOBAL_ATOMIC_ADD_F32 per lane into acc[dst].
// ---------------------------------------------------------------------------
__global__ __launch_bounds__(256) void gat_edge_scatter(
    const int* __restrict__ esrc, const int* __restrict__ edst,
    const float* __restrict__ aS, const float* __restrict__ aD,
    const float* __restrict__ mmax, const float* __restrict__ denom,
    const float* __restrict__ xp, float* __restrict__ acc, int E, int N)
{
    const int wave = (blockIdx.x * blockDim.x + threadIdx.x) >> 5;
    const int lane = threadIdx.x & 31;
    if (wave >= E + N) return;
    int s, d;
    if (wave < E) { s = esrc[wave]; d = edst[wave]; }
    else          { s = d = wave - E; }
    const int head = lane >> 4;
    float e = lrelu(aS[s * 2 + head] + aD[d * 2 + head]);
    const float alpha = __expf(e - mmax[d * 2 + head]) /
                        (denom[d * 2 + head] + SM_EPS);
    const int fo = lane * 4;                       // lanes cover all 128 feats
    float4 v = *(const float4*)(xp + (size_t)s * XDIM + fo);
    float* ap = acc + (size_t)d * XDIM + fo;
    atomicAdd(ap + 0, v.x * alpha);
    atomicAdd(ap + 1, v.y * alpha);
    atomicAdd(ap + 2, v.z * alpha);
    atomicAdd(ap + 3, v.w * alpha);
}

// ---------------------------------------------------------------------------
// Kernel 6: mean over heads + bias + LayerNorm(64). One wave per node,
// 2 features per lane, wave32 butterfly reductions.
// ---------------------------------------------------------------------------
__global__ __launch_bounds__(256) void gat_finalize(
    const float* __restrict__ acc, const float* __restrict__ bias,
    const float* __restrict__ gamma, const float* __restrict__ beta,
    float* __restrict__ out, int N)
{
    const int wave = (blockIdx.x * blockDim.x + threadIdx.x) >> 5;
    const int lane = threadIdx.x & 31;
    if (wave >= N) return;
    const float* a = acc + (size_t)wave * XDIM;
    const int f0 = lane, f1 = lane + 32;
    float v0 = (a[f0] + a[OUT_F + f0]) * 0.5f + bias[f0];
    float v1 = (a[f1] + a[OUT_F + f1]) * 0.5f + bias[f1];

    float sum = v0 + v1;
    #pragma unroll
    for (int off = 16; off >= 1; off >>= 1) sum += __shfl_xor(sum, off, 32);
    const float mu = sum * (1.0f / OUT_F);

    const float d0 = v0 - mu, d1 = v1 - mu;
    float sq = d0 * d0 + d1 * d1;
    #pragma unroll
    for (int off = 16; off >= 1; off >>= 1) sq += __shfl_xor(sq, off, 32);
    const float inv = rsqrtf(sq * (1.0f / OUT_F) + LN_EPS);

    out[(size_t)wave * OUT_F + f0] = d0 * inv * gamma[f0] + beta[f0];
    out[(size_t)wave * OUT_F + f1] = d1 * inv * gamma[f1] + beta[f1];
}

// ---------------------------------------------------------------------------
extern "C" void kernel_launch(void* const* d_in, const int* in_sizes, int n_in,
                              void* d_out, int out_size, void* d_ws, size_t ws_size,
                              hipStream_t stream)
{
    const float* X    = (const float*)d_in[0];   // [N,128]
    const int*   ei   = (const int*)  d_in[1];   // [2,E]
    const float* W    = (const float*)d_in[2];   // [128,128]
    const float* attS = (const float*)d_in[3];   // [2,64]
    const float* attD = (const float*)d_in[4];   // [2,64]
    const float* bias = (const float*)d_in[5];   // [64]
    const float* g    = (const float*)d_in[6];   // [64]
    const float* b    = (const float*)d_in[7];   // [64]
    float* out = (float*)d_out;

    const int N = in_sizes[0] / IN_F;
    const int E = in_sizes[1] / 2;
    const int* esrc = ei;
    const int* edst = ei + E;
    const int Et = E + N;

    // workspace layout (floats): xp[N*128] | acc[N*128] | aS[2N] | aD[2N]
    //                            | m[2N] | denom[2N]
    float* ws    = (float*)d_ws;
    float* xp    = ws;
    float* accb  = xp   + (size_t)N * XDIM;
    float* aS    = accb + (size_t)N * XDIM;
    float* aD    = aS   + (size_t)N * 2;
    float* mmax  = aD   + (size_t)N * 2;
    float* denom = mmax + (size_t)N * 2;

    // 1) projection GEMM (WMMA fp32): one wave per 16-row strip
    {
        const int waves = (N + 15) / 16;
        gat_gemm_wmma<<<(waves + 7) / 8, 256, 0, stream>>>(X, W, xp, N);
    }
    // 2) logits + init
    gat_logits_init<<<(N + 7) / 8, 256, 0, stream>>>(
        xp, attS, attD, aS, aD, mmax, denom, accb, N);
    // 3) segment max
    gat_edge_max<<<(Et + 255) / 256, 256, 0, stream>>>(
        esrc, edst, aS, aD, mmax, E, N);
    // 4) segment sum of exp
    gat_edge_sum<<<(Et + 255) / 256, 256, 0, stream>>>(
        esrc, edst, aS, aD, mmax, denom, E, N);
    // 5) weighted scatter-add
    gat_edge_scatter<<<(Et + 7) / 8, 256, 0, stream>>>(
        esrc, edst, aS, aD, mmax, denom, xp, accb, E, N);
    // 6) head mean + LayerNorm
    gat_finalize<<<(N + 7) / 8, 256, 0, stream>>>(accb, bias, g, b, out, N);
}